// Model_2layers_45157286150379
// MI455X (gfx1250) — compile-verified
//
#include <hip/hip_runtime.h>
#include <math.h>

// ---------------- problem constants ----------------
#define D_IN   40
#define HID    250
#define HP     252          // padded hidden stride (mult of 4 -> aligned float2/float4)
#define BATCH  256
#define SEQ    512
#define NWG    64
#define TPB    128          // 4 wave32/WG; WG covers ONE batch tile x 4 hidden tiles

typedef float v2f __attribute__((ext_vector_type(2)));
typedef float v8f __attribute__((ext_vector_type(8)));

// ---------------- WMMA fp32 16x16x4 ----------------
__device__ __forceinline__ v8f wmma4(v2f a, v2f b, v8f c) {
    return __builtin_amdgcn_wmma_f32_16x16x4_f32(
        false, a, false, b, (short)0, c, false, false);
}

__device__ __forceinline__ float sigm(float x) { return 1.0f / (1.0f + expf(-x)); }

__device__ __forceinline__ v8f splat8(float v) {
    v8f r;
#pragma unroll
    for (int i = 0; i < 8; ++i) r[i] = v;
    return r;
}

// ---------------- CDNA5 async global->LDS staging ----------------
__device__ __forceinline__ unsigned lds_addr(const void* p) {
    return (unsigned)(unsigned long long)
           (const __attribute__((address_space(3))) char*)p;
}
__device__ __forceinline__ void async_copy_b128(const void* gsrc, void* ldst) {
    unsigned l = lds_addr(ldst);
    asm volatile("global_load_async_to_lds_b128 %0, %1, off"
                 :: "v"(l), "v"(gsrc) : "memory");
}
__device__ __forceinline__ void wait_async0() {
    asm volatile("s_wait_asynccnt 0x0" ::: "memory");
}

// Stage a ROWS x (NF4*4) float tile (16B-aligned rows) global -> LDS, whole WG.
template <int NF4, int ROWS>
__device__ __forceinline__ void stage_tile(float* sh, int sh_stride,
                                           const float* g, int g_stride) {
    for (int i = threadIdx.x; i < NF4 * ROWS; i += TPB) {
        const int r = i / NF4, c = (i - r * NF4) * 4;
        async_copy_b128(g + (size_t)r * g_stride + c, sh + r * sh_stride + c);
    }
}

// ---------------- grid-wide barrier (generation counter) ----------------
__device__ __forceinline__ void grid_sync(int* cnt, int* gen, int nwg) {
    __syncthreads();
    if (threadIdx.x == 0) {
        __threadfence();
        int g = __hip_atomic_load(gen, __ATOMIC_ACQUIRE, __HIP_MEMORY_SCOPE_AGENT);
        int prev = __hip_atomic_fetch_add(cnt, 1, __ATOMIC_ACQ_REL, __HIP_MEMORY_SCOPE_AGENT);
        if (prev == nwg - 1) {
            __hip_atomic_store(cnt, 0, __ATOMIC_RELAXED, __HIP_MEMORY_SCOPE_AGENT);
            __hip_atomic_fetch_add(gen, 1, __ATOMIC_ACQ_REL, __HIP_MEMORY_SCOPE_AGENT);
        } else {
            while (__hip_atomic_load(gen, __ATOMIC_ACQUIRE, __HIP_MEMORY_SCOPE_AGENT) == g) {
                __builtin_amdgcn_s_sleep(2);
            }
        }
        __threadfence();
    }
    __syncthreads();
}

// ---------------- acc[4 gates] += A_tile(LDS) * W^T, B software-pipelined ----------
// arow: LDS row base for this lane's M row. Tail relies on LDS pad cols (>=250)
// being zero so the upper-lane-half k=250/251 slots contribute 0*B.
__device__ __forceinline__ void mm_lds(v8f acc[4], const float* arow,
                                       const float* __restrict__ W, int ldw,
                                       int kfull, int has_tail,
                                       int koff, int hcolc) {
    const float* wr0 = W + (size_t)(0 * HID + hcolc) * ldw;
    const float* wr1 = W + (size_t)(1 * HID + hcolc) * ldw;
    const float* wr2 = W + (size_t)(2 * HID + hcolc) * ldw;
    const float* wr3 = W + (size_t)(3 * HID + hcolc) * ldw;

    int kc = 0;
    for (; kc + 4 <= kfull; kc += 4) {              // 4 K-chunks per stage: 16 b64 loads
        v2f bf[16];
#pragma unroll
        for (int u = 0; u < 4; ++u) {
            const int k0 = (kc + u) * 4 + koff;
            bf[4 * u + 0] = *(const v2f*)(wr0 + k0);
            bf[4 * u + 1] = *(const v2f*)(wr1 + k0);
            bf[4 * u + 2] = *(const v2f*)(wr2 + k0);
            bf[4 * u + 3] = *(const v2f*)(wr3 + k0);
        }
#pragma unroll
        for (int u = 0; u < 4; ++u) {
            v2f a = *(const v2f*)(arow + (kc + u) * 4 + koff);
            acc[0] = wmma4(a, bf[4 * u + 0], acc[0]);
            acc[1] = wmma4(a, bf[4 * u + 1], acc[1]);
            acc[2] = wmma4(a, bf[4 * u + 2], acc[2]);
            acc[3] = wmma4(a, bf[4 * u + 3], acc[3]);
        }
    }
    for (; kc < kfull; ++kc) {
        const int k0 = kc * 4 + koff;
        v2f a = *(const v2f*)(arow + k0);
        acc[0] = wmma4(a, *(const v2f*)(wr0 + k0), acc[0]);
        acc[1] = wmma4(a, *(const v2f*)(wr1 + k0), acc[1]);
        acc[2] = wmma4(a, *(const v2f*)(wr2 + k0), acc[2]);
        acc[3] = wmma4(a, *(const v2f*)(wr3 + k0), acc[3]);
    }
    if (has_tail) {
        const int k0 = kfull * 4;                   // 248; valid k: 248,249
        v2f a = *(const v2f*)(arow + k0 + koff);    // upper half reads zero pad
        acc[0] = wmma4(a, *(const v2f*)(wr0 + k0), acc[0]);
        acc[1] = wmma4(a, *(const v2f*)(wr1 + k0), acc[1]);
        acc[2] = wmma4(a, *(const v2f*)(wr2 + k0), acc[2]);
        acc[3] = wmma4(a, *(const v2f*)(wr3 + k0), acc[3]);
    }
}

// ---------------- gate math + register-resident cell update ----------------
__device__ __forceinline__ void gates_update(const v8f acc[4], v8f* creg,
                                             float* __restrict__ hdst,
                                             int brow0, int hcol, bool hvalid) {
#pragma unroll
    for (int r = 0; r < 8; ++r) {
        float ig = sigm(acc[0][r]);
        float fg = sigm(acc[1][r]);
        float gg = tanhf(acc[2][r]);
        float og = sigm(acc[3][r]);
        float c  = fg * (*creg)[r] + ig * gg;
        (*creg)[r] = c;
        float h  = og * tanhf(c);
        if (hvalid) hdst[(size_t)(brow0 + r) * HP + hcol] = h;
    }
}

__global__ __launch_bounds__(TPB, 1)
void lstm2_persistent(const float* __restrict__ x,
                      const float* __restrict__ Wih0, const float* __restrict__ Whh0,
                      const float* __restrict__ bih0, const float* __restrict__ bhh0,
                      const float* __restrict__ Wih1, const float* __restrict__ Whh1,
                      const float* __restrict__ bih1, const float* __restrict__ bhh1,
                      const float* __restrict__ Wlin, const float* __restrict__ blin,
                      float* __restrict__ out,
                      float* __restrict__ h1buf,   // [2][BATCH][HP], zeroed
                      float* __restrict__ h2buf,   // [2][BATCH][HP], zeroed
                      int* __restrict__ bar)       // [2] ints, zeroed
{
    __shared__ __align__(16) float shA0[16 * HP];   // A tile (h), 16.1KB
    __shared__ __align__(16) float shA1[16 * HP];   // second A tile for layer 2
    __shared__ __align__(16) float shX [16 * D_IN]; // x tile

    const int lane = threadIdx.x & 31;
    const int wid  = threadIdx.x >> 5;
    const int bt = blockIdx.x >> 2;                       // uniform batch tile per WG
    const int ht = ((blockIdx.x & 3) << 2) | wid;         // hidden tile 0..15
    const int b0 = bt * 16;
    const int n0 = ht * 16;
    const int lm = lane & 15;
    const int lh = lane >> 4;
    const int koff = lh * 2;
    const int hcol  = n0 + lm;
    const int hcolc = hcol < HID ? hcol : (HID - 1);
    const bool hvalid = hcol < HID;
    const int brow0 = b0 + lh * 8;
    const int wave = blockIdx.x * 4 + wid;                // 0..255 (waves 0..7 do head)

    v8f bias1[4], bias2[4];
#pragma unroll
    for (int g = 0; g < 4; ++g) {
        bias1[g] = splat8(bih0[g * HID + hcolc] + bhh0[g * HID + hcolc]);
        bias2[g] = splat8(bih1[g * HID + hcolc] + bhh1[g * HID + hcolc]);
    }
    v8f c1 = splat8(0.0f), c2 = splat8(0.0f);

    for (int t = 0; t < SEQ; ++t) {
        const int cur = t & 1, nxt = cur ^ 1;
        const float* h1c = h1buf + (size_t)cur * BATCH * HP;
        float*       h1n = h1buf + (size_t)nxt * BATCH * HP;
        const float* h2c = h2buf + (size_t)cur * BATCH * HP;
        float*       h2n = h2buf + (size_t)nxt * BATCH * HP;

        // ---- head output for t-1 (h2buf[cur] was published before last barrier) ----
        if (t > 0 && wave < 8) {
            const int b = wave * 32 + lane;
            const float* hrow = h2c + (size_t)b * HP;
            float s = blin[0];
            for (int k = 0; k < HID; ++k) s += hrow[k] * Wlin[k];
            out[(size_t)b * SEQ + (t - 1)] = sigm(s);
        }

        // ---- stage layer-1 A tiles (shared by the whole WG) via async->LDS ----
        stage_tile<HP / 4, 16>(shA0, HP, h1c + (size_t)b0 * HP, HP);
        stage_tile<D_IN / 4, 16>(shX, D_IN, x + ((size_t)b0 * SEQ + t) * D_IN, SEQ * D_IN);
        wait_async0();
        __syncthreads();

        // prefetch next timestep's x tile rows
        if (t + 1 < SEQ && threadIdx.x < 16) {
            __builtin_prefetch(x + ((size_t)(b0 + threadIdx.x) * SEQ + (t + 1)) * D_IN, 0, 1);
        }

        // ---- layer 1: g = x*Wih0^T + h1*Whh0^T + bias ----
        {
            v8f acc[4] = { bias1[0], bias1[1], bias1[2], bias1[3] };
            mm_lds(acc, shX  + (size_t)lm * D_IN, Wih0, D_IN, D_IN / 4, 0, koff, hcolc);
            mm_lds(acc, shA0 + (size_t)lm * HP,   Whh0, HID, HID / 4,  1, koff, hcolc);
            gates_update(acc, &c1, h1n, brow0, hcol, hvalid);
        }
        grid_sync(&bar[0], &bar[1], NWG);

        // ---- stage layer-2 A tiles: h1_new and h2_cur ----
        stage_tile<HP / 4, 16>(shA0, HP, h1n + (size_t)b0 * HP, HP);
        stage_tile<HP / 4, 16>(shA1, HP, h2c + (size_t)b0 * HP, HP);
        wait_async0();
        __syncthreads();

        // ---- layer 2: g = h1_new*Wih1^T + h2*Whh1^T + bias ----
        {
            v8f acc[4] = { bias2[0], bias2[1], bias2[2], bias2[3] };
            mm_lds(acc, shA0 + (size_t)lm * HP, Wih1, HID, HID / 4, 1, koff, hcolc);
            mm_lds(acc, shA1 + (size_t)lm * HP, Whh1, HID, HID / 4, 1, koff, hcolc);
            gates_update(acc, &c2, h2n, brow0, hcol, hvalid);
        }
        grid_sync(&bar[0], &bar[1], NWG);
    }

    // ---- final head output for t = SEQ-1 (h2 ends in buffer SEQ&1 == 0) ----
    if (wave < 8) {
        const int b = wave * 32 + lane;
        const float* hrow = h2buf + (size_t)(SEQ & 1) * BATCH * HP + (size_t)b * HP;
        float s = blin[0];
        for (int k = 0; k < HID; ++k) s += hrow[k] * Wlin[k];
        out[(size_t)b * SEQ + (SEQ - 1)] = sigm(s);
    }
}

extern "C" void kernel_launch(void* const* d_in, const int* in_sizes, int n_in,
                              void* d_out, int out_size, void* d_ws, size_t ws_size,
                              hipStream_t stream) {
    (void)in_sizes; (void)n_in; (void)out_size; (void)ws_size;
    const float* x    = (const float*)d_in[0];
    const float* Wih0 = (const float*)d_in[1];
    const float* Whh0 = (const float*)d_in[2];
    const float* bih0 = (const float*)d_in[3];
    const float* bhh0 = (const float*)d_in[4];
    const float* Wih1 = (const float*)d_in[5];
    const float* Whh1 = (const float*)d_in[6];
    const float* bih1 = (const float*)d_in[7];
    const float* bhh1 = (const float*)d_in[8];
    const float* Wlin = (const float*)d_in[9];
    const float* blin = (const float*)d_in[10];

    float* ws = (float*)d_ws;
    float* h1buf = ws;                               // 2*BATCH*HP floats
    float* h2buf = ws + 2 * BATCH * HP;              // 2*BATCH*HP floats
    int*   bar   = (int*)(ws + 4 * BATCH * HP);      // 2 ints

    const size_t clear_bytes = (size_t)4 * BATCH * HP * sizeof(float) + 2 * sizeof(int);
    hipMemsetAsync(d_ws, 0, clear_bytes, stream);    // zero h state (+pads) and barrier

    lstm2_persistent<<<NWG, TPB, 0, stream>>>(
        x, Wih0, Whh0, bih0, bhh0, Wih1, Whh1, bih1, bhh1, Wlin, blin,
        (float*)d_out, h1buf, h2buf, bar);
}